// LossSegmentwiseFreeTags_20847771255017
// MI455X (gfx1250) — compile-verified
//
#include <hip/hip_runtime.h>
#include <hip/hip_bf16.h>

#define B_      4
#define C_      32          // S*D
#define P_      (512*512)
#define NSEG_   64
#define TILE_PIX 64
#define CHUNK1  2048        // pixels per block, kernel 1 (512 blocks -> 2/WGP)
#define CHUNK3  2048        // pixels per block, kernel 3
#define LSTRIDE 66          // LDS tile row stride (even -> b64, bank-clean)
#define USE_ASYNC_LDS 1     // CDNA5 global_load_async_to_lds_b64 staging path

typedef __attribute__((ext_vector_type(2))) float v2f;
typedef __attribute__((ext_vector_type(8))) float v8f;

__device__ __forceinline__ void wait_async0() {
#if defined(__has_builtin)
#if __has_builtin(__builtin_amdgcn_s_wait_asynccnt)
    __builtin_amdgcn_s_wait_asynccnt(0);
#else
    asm volatile("s_wait_asynccnt 0" ::: "memory");
#endif
#else
    asm volatile("s_wait_asynccnt 0" ::: "memory");
#endif
}

// async DMA: copies 8 bytes per lane from global sbase + voff to the LDS address in ldsoff, ASYNCcnt-tracked
__device__ __forceinline__ void async_copy_b64(uint32_t ldsoff, uint32_t voff,
                                               const void* sbase) {
    asm volatile("global_load_async_to_lds_b64 %0, %1, %2"
                 :: "v"(ldsoff), "v"(voff), "s"((uint64_t)(uintptr_t)sbase)
                 : "memory");
}

// ---------------------------------------------------------------- zero scratch
__global__ void zero_kernel(float* p, int n) {
    int i = blockIdx.x * 256 + threadIdx.x;
    if (i < n) p[i] = 0.0f;
}

// ---------------------------------------------------------------- kernel 1:
// segment sums via one-hot WMMA: sums[b][seg][c] = sum_{p: lab==seg} t[c][p]
__global__ __launch_bounds__(256) void seg_sum_kernel(
        const float* __restrict__ preds,   // [B][32][P]
        const int*   __restrict__ labels,  // [B][P]
        float* __restrict__ sums,          // [B][64][32]
        float* __restrict__ counts)        // [B][64]
{
    __shared__ float tile[C_ * LSTRIDE];   // 32 x 66 staged tags
    __shared__ int   lab[TILE_PIX];
    __shared__ float merge[NSEG_ * 33];    // padded merge array
    __shared__ float cnt_l[NSEG_];

    const int tid  = threadIdx.x;
    const int lane = tid & 31;
    const int wv   = tid >> 5;             // 0..7 (8 waves)
    const int m    = lane & 15;
    const int half = lane >> 4;            // 0/1

    const int nchunk = P_ / CHUNK1;
    const int b  = blockIdx.x / nchunk;
    const int ch = blockIdx.x % nchunk;
    const long pbase = (long)ch * CHUNK1;
    const float* pb = preds  + (long)b * C_ * P_;
    const int*   lb = labels + (long)b * P_;

    if (tid < NSEG_) cnt_l[tid] = 0.0f;

    v8f acc[4][2];                         // [seg-tile][chan-tile] = 64x32 in VGPRs
    #pragma unroll
    for (int st = 0; st < 4; ++st)
        #pragma unroll
        for (int ct = 0; ct < 2; ++ct)
            acc[st][ct] = (v8f){0,0,0,0,0,0,0,0};

    for (int t0 = 0; t0 < CHUNK1; t0 += TILE_PIX) {
        const long tb = pbase + t0;
        if (tid < TILE_PIX) lab[tid] = lb[tb + tid];
        // stage 32x64 tag tile: 1024 x b64, 4 per thread, coalesced
        const float* gbase = pb + tb;      // uniform -> SGPR pair for GVS mode
        #pragma unroll
        for (int it = 0; it < 4; ++it) {
            int f  = it * 256 + tid;       // float2 index: 32 per row
            int c  = f >> 5;
            int p2 = f & 31;
            float* d = &tile[c * LSTRIDE + p2 * 2];
#if USE_ASYNC_LDS
            // direct global->LDS DMA (no VGPR round-trip), tracked by ASYNCcnt
            uint32_t voff = (uint32_t)(((long)c * P_ + p2 * 2) * sizeof(float));
            async_copy_b64((uint32_t)(uintptr_t)d, voff, gbase);
#else
            const float* g = gbase + (long)c * P_ + p2 * 2;
            __builtin_prefetch(g + TILE_PIX, 0, 1);
            float2 v = *(const float2*)g;
            d[0] = v.x; d[1] = v.y;
#endif
        }
#if USE_ASYNC_LDS
        wait_async0();
#endif
        __syncthreads();

        if (tid < TILE_PIX) atomicAdd(&cnt_l[lab[tid]], 1.0f);

        // each wave: 8 pixels -> 2 K-steps of 4; 8 WMMAs per K-step
        #pragma unroll
        for (int step = 0; step < 2; ++step) {
            const int kb = wv * 8 + step * 4;
            const int s0 = lab[kb + 2 * half + 0];
            const int s1 = lab[kb + 2 * half + 1];
            v2f bm[2];
            #pragma unroll
            for (int ct = 0; ct < 2; ++ct) {
                const float* src = &tile[(ct * 16 + m) * LSTRIDE + kb + 2 * half];
                bm[ct].x = src[0];
                bm[ct].y = src[1];
            }
            #pragma unroll
            for (int st = 0; st < 4; ++st) {
                const int off = st * 16 + m;
                v2f a;                       // one-hot A tile (exact in f32)
                a.x = (s0 == off) ? 1.0f : 0.0f;
                a.y = (s1 == off) ? 1.0f : 0.0f;
                #pragma unroll
                for (int ct = 0; ct < 2; ++ct) {
                    acc[st][ct] = __builtin_amdgcn_wmma_f32_16x16x4_f32(
                        false, a, false, bm[ct], (short)0, acc[st][ct], false, false);
                }
            }
        }
        __syncthreads();
    }

    // merge 8 per-wave accumulators through LDS, then one global-atomic batch
    for (int i = tid; i < NSEG_ * 33; i += 256) merge[i] = 0.0f;
    __syncthreads();
    #pragma unroll
    for (int st = 0; st < 4; ++st)
        #pragma unroll
        for (int ct = 0; ct < 2; ++ct)
            #pragma unroll
            for (int v = 0; v < 8; ++v) {
                int seg = st * 16 + v + 8 * half;   // C/D layout: M = v + 8*half
                int c   = ct * 16 + m;              // N = lane&15
                atomicAdd(&merge[seg * 33 + c], acc[st][ct][v]);
            }
    __syncthreads();
    for (int i = tid; i < NSEG_ * C_; i += 256) {
        int seg = i >> 5, c = i & 31;
        atomicAdd(&sums[(b * NSEG_ + seg) * C_ + c], merge[seg * 33 + c]);
    }
    if (tid < NSEG_) atomicAdd(&counts[b * NSEG_ + tid], cnt_l[tid]);
}

// ---------------------------------------------------------------- kernel 2:
// centroids + pairwise push loss (tiny)
__global__ __launch_bounds__(256) void centroid_push_kernel(
        const float* __restrict__ sums, const float* __restrict__ counts,
        float* __restrict__ cent, float* __restrict__ push)
{
    __shared__ float lc[NSEG_ * 33];
    __shared__ float red;
    const int b = blockIdx.x;
    const int tid = threadIdx.x;
    for (int i = tid; i < NSEG_ * C_; i += 256) {
        int seg = i >> 5, c = i & 31;
        float cv = sums[b * 2048 + i] / fmaxf(counts[b * 64 + seg], 1.0f);
        cent[b * 2048 + i] = cv;
        lc[seg * 33 + c] = cv;
    }
    if (tid == 0) red = 0.0f;
    __syncthreads();
    float local = 0.0f;
    for (int idx = tid; idx < 2016; idx += 256) {
        int i = 0, rem = idx;
        while (rem >= 63 - i) { rem -= 63 - i; ++i; }
        int j = i + 1 + rem;
        #pragma unroll
        for (int s = 0; s < 2; ++s) {
            float dsum = 0.0f;
            #pragma unroll
            for (int d = 0; d < 16; ++d)
                dsum += fabsf(lc[i * 33 + s * 16 + d] - lc[j * 33 + s * 16 + d]);
            float r = fmaxf(0.25f - dsum * (1.0f / 16.0f), 0.0f);
            local += r * r;
        }
    }
    atomicAdd(&red, local);
    __syncthreads();
    if (tid == 0) push[b] = red;
}

// ---------------------------------------------------------------- kernel 3:
// pull loss; preds re-read (L2-resident), centroids cached in LDS
__global__ __launch_bounds__(256) void pull_kernel(
        const float* __restrict__ preds, const int* __restrict__ labels,
        const float* __restrict__ cent, float* __restrict__ pull)
{
    __shared__ float lc[NSEG_ * 33];
    __shared__ float red;
    const int tid = threadIdx.x;
    const int nchunk = P_ / CHUNK3;
    const int b  = blockIdx.x / nchunk;
    const int ch = blockIdx.x % nchunk;
    for (int i = tid; i < NSEG_ * C_; i += 256)
        lc[(i >> 5) * 33 + (i & 31)] = cent[b * 2048 + i];
    if (tid == 0) red = 0.0f;
    __syncthreads();

    const float* pb = preds  + (long)b * C_ * P_;
    const int*   lb = labels + (long)b * P_;
    float local = 0.0f;
    const int pend = ch * CHUNK3 + CHUNK3;
    for (int p0 = ch * CHUNK3; p0 < pend; p0 += 256) {
        const int p = p0 + tid;
        const int seg = lb[p];
        __builtin_prefetch(pb + p + 256, 0, 1);
        float a0 = 0.0f, a1 = 0.0f;
        #pragma unroll
        for (int c = 0; c < 16; ++c)
            a0 += fabsf(pb[(long)c * P_ + p] - lc[seg * 33 + c]);
        #pragma unroll
        for (int c = 16; c < 32; ++c)
            a1 += fabsf(pb[(long)c * P_ + p] - lc[seg * 33 + c]);
        a0 *= (1.0f / 16.0f);
        a1 *= (1.0f / 16.0f);
        local += a0 * a0 + a1 * a1;
    }
    atomicAdd(&red, local);
    __syncthreads();
    if (tid == 0) atomicAdd(&pull[b], red);
}

// ---------------------------------------------------------------- kernel 4
__global__ void finalize_kernel(const float* push, const float* pull, float* out) {
    float acc = 0.0f;
    for (int b = 0; b < B_; ++b) {
        float pu = push[b] * (1.0f / (2.0f * 2016.0f));
        float pl = pull[b] * (1.0f / (2.0f * (float)P_));
        acc += 1.0f * pu + 0.1f * pl;
    }
    out[0] = acc * (1.0f / (float)B_);
}

// ---------------------------------------------------------------- launcher
extern "C" void kernel_launch(void* const* d_in, const int* in_sizes, int n_in,
                              void* d_out, int out_size, void* d_ws, size_t ws_size,
                              hipStream_t stream) {
    const float* preds  = (const float*)d_in[0];   // (4,2,16,512,512) f32
    const int*   labels = (const int*)d_in[1];     // (4,1,512,512) i32

    float* ws     = (float*)d_ws;
    float* sums   = ws;              // 4*64*32 = 8192
    float* counts = ws + 8192;       // 4*64    = 256
    float* push   = ws + 8448;       // 4
    float* pull   = ws + 8452;       // 4
    float* cent   = ws + 8456;       // 8192 (fully overwritten, no zeroing)

    zero_kernel<<<(8456 + 255) / 256, 256, 0, stream>>>(ws, 8456);
    seg_sum_kernel<<<B_ * (P_ / CHUNK1), 256, 0, stream>>>(preds, labels, sums, counts);
    centroid_push_kernel<<<B_, 256, 0, stream>>>(sums, counts, cent, push);
    pull_kernel<<<B_ * (P_ / CHUNK3), 256, 0, stream>>>(preds, labels, cent, pull);
    finalize_kernel<<<1, 1, 0, stream>>>(push, pull, (float*)d_out);
}